// dBGEmbedding_37091337568554
// MI455X (gfx1250) — compile-verified
//
#include <hip/hip_runtime.h>
#include <hip/hip_bf16.h>

#define B_   64
#define L_   512
#define NF_  1024
#define F_   16
#define D_   512

typedef __attribute__((ext_vector_type(2))) float v2f;
typedef __attribute__((ext_vector_type(8))) float v8f;
typedef __attribute__((ext_vector_type(2))) short s2;

// ---------------------------------------------------------------------------
// Kernel 1: discretize -> tokens, centered tokens rc[b,l], ||rc||_2 per row
// ---------------------------------------------------------------------------
__global__ void k_tokens(const float* __restrict__ x, int* __restrict__ tokens,
                         float* __restrict__ rc, float* __restrict__ norm_rc) {
    const int b = blockIdx.x, tid = threadIdx.x;           // 256 threads
    __shared__ float red[256];
    __shared__ int   tok_s[L_];

    float sum = 0.f;
    for (int l = tid; l < L_; l += 256) {
        float v = x[b * L_ + l];
        float t = floorf((v + 3.0f) / 6.0f * 10.0f);
        t = fminf(fmaxf(t, 0.0f), 9.0f);
        int ti = (int)t;
        tok_s[l] = ti;
        tokens[b * L_ + l] = ti;
        sum += t;
    }
    red[tid] = sum; __syncthreads();
    for (int s = 128; s > 0; s >>= 1) { if (tid < s) red[tid] += red[tid + s]; __syncthreads(); }
    const float mean = red[0] * (1.0f / (float)L_);
    __syncthreads();

    float ss = 0.f;
    for (int l = tid; l < L_; l += 256) {
        float r = (float)tok_s[l] - mean;
        rc[b * L_ + l] = r;
        ss += r * r;
    }
    red[tid] = ss; __syncthreads();
    for (int s = 128; s > 0; s >>= 1) { if (tid < s) red[tid] += red[tid + s]; __syncthreads(); }
    if (tid == 0) norm_rc[b] = sqrtf(red[0]);
}

// ---------------------------------------------------------------------------
// Kernel 2: linear-interp feature rows 16 -> 512, int-truncate, center,
//           store fc[n,l] and ||fc||_2 per row
// ---------------------------------------------------------------------------
__global__ void k_fc(const int* __restrict__ features, float* __restrict__ fc,
                     float* __restrict__ norm_fc) {
    const int n = blockIdx.x, tid = threadIdx.x;           // 256 threads
    __shared__ float fr[F_];
    __shared__ float red[256];

    if (tid < F_) fr[tid] = (float)features[n * F_ + tid];
    __syncthreads();

    float fiv[2];
    float sum = 0.f;
#pragma unroll
    for (int i = 0; i < 2; i++) {
        int l = tid + i * 256;
        float p = (float)l * (15.0f / 511.0f);
        int k = (int)p; if (k > 14) k = 14;
        float frac = p - (float)k;
        float v = fr[k] + frac * (fr[k + 1] - fr[k]);
        float fi = (float)(int)v;                          // astype(int) in reference
        fiv[i] = fi;
        sum += fi;
    }
    red[tid] = sum; __syncthreads();
    for (int s = 128; s > 0; s >>= 1) { if (tid < s) red[tid] += red[tid + s]; __syncthreads(); }
    const float mean = red[0] * (1.0f / (float)L_);
    __syncthreads();

    float ss = 0.f;
#pragma unroll
    for (int i = 0; i < 2; i++) {
        int l = tid + i * 256;
        float c = fiv[i] - mean;
        fc[n * L_ + l] = c;
        ss += c * c;
    }
    red[tid] = ss; __syncthreads();
    for (int s = 128; s > 0; s >>= 1) { if (tid < s) red[tid] += red[tid + s]; __syncthreads(); }
    if (tid == 0) norm_fc[n] = sqrtf(red[0]);
}

// ---------------------------------------------------------------------------
// Kernel 3: Levenshtein DP, two features per thread packed in short2 so the
//           DP runs on v_pk_add_i16 / v_pk_min_i16. DP values <= 528 fit i16.
// ---------------------------------------------------------------------------
__global__ void k_lev(const int* __restrict__ tokens, const int* __restrict__ features,
                      float* __restrict__ dist) {
    const int b   = blockIdx.y;
    const int nfA = blockIdx.x * 256 + threadIdx.x;        // [0,512)
    const int nfB = nfA + 512;

    __shared__ int tok_s[L_];
    for (int l = threadIdx.x; l < L_; l += 256) tok_s[l] = tokens[b * L_ + l];
    __syncthreads();

    s2 f2[F_];
#pragma unroll
    for (int j = 0; j < F_; j++) {
        s2 t; t.x = (short)features[nfA * F_ + j]; t.y = (short)features[nfB * F_ + j];
        f2[j] = t;
    }

    s2 prev[F_ + 1];
#pragma unroll
    for (int j = 0; j <= F_; j++) { prev[j].x = (short)j; prev[j].y = (short)j; }

    const s2 one = {1, 1};
    for (int l = 0; l < L_; l++) {
        short t = (short)tok_s[l];
        s2 t2; t2.x = t; t2.y = t;
        s2 diag = prev[0];
        s2 cur0; cur0.x = (short)(l + 1); cur0.y = (short)(l + 1);
        prev[0] = cur0;
        s2 left = cur0;
#pragma unroll
        for (int j = 1; j <= F_; j++) {
            s2 old  = prev[j];
            s2 neq  = (t2 != f2[j - 1]);                   // -1 / 0 per lane
            s2 sub  = diag + (neq & one);
            s2 v    = __builtin_elementwise_min(
                          __builtin_elementwise_min(old + one, left + one), sub);
            prev[j] = v; left = v; diag = old;
        }
    }
    dist[b * NF_ + nfA] = (float)prev[F_].x;
    dist[b * NF_ + nfB] = (float)prev[F_].y;
}

// ---------------------------------------------------------------------------
// Kernel 4: per-row normalize distances (two-pass mean/var, unbiased=False)
// ---------------------------------------------------------------------------
__global__ void k_dbg(const float* __restrict__ dist, float* __restrict__ dbg) {
    const int b = blockIdx.x, tid = threadIdx.x;           // 256 threads
    __shared__ float red[256];

    float s = 0.f;
    for (int n = tid; n < NF_; n += 256) s += dist[b * NF_ + n];
    red[tid] = s; __syncthreads();
    for (int st = 128; st > 0; st >>= 1) { if (tid < st) red[tid] += red[tid + st]; __syncthreads(); }
    const float m = red[0] * (1.0f / (float)NF_);
    __syncthreads();

    float sq = 0.f;
    for (int n = tid; n < NF_; n += 256) { float d = dist[b * NF_ + n] - m; sq += d * d; }
    red[tid] = sq; __syncthreads();
    for (int st = 128; st > 0; st >>= 1) { if (tid < st) red[tid] += red[tid + st]; __syncthreads(); }
    const float inv = rsqrtf(red[0] * (1.0f / (float)NF_) + 1e-5f);

    for (int n = tid; n < NF_; n += 256)
        dbg[b * NF_ + n] = (dist[b * NF_ + n] - m) * inv;
}

// ---------------------------------------------------------------------------
// Kernel 5: corr[64,1024] = rc[64,512] x fc^T[512,1024] via
//           V_WMMA_F32_16X16X4_F32 (full f32), then Pearson normalization.
// One wave per 16x16 output tile. A: lane=(M=lane&15), K=vgpr+2*(lane>>4);
// B: lane=(N=lane&15), same K pattern -> both are float2 loads from row-major.
// D: M = r + 8*(lane>>4), N = lane&15.
// ---------------------------------------------------------------------------
__global__ void k_corr(const float* __restrict__ rc, const float* __restrict__ fc,
                       const float* __restrict__ norm_rc, const float* __restrict__ norm_fc,
                       float* __restrict__ corr) {
    const int bt = blockIdx.x;                             // 0..3   (M tiles)
    const int nt = blockIdx.y;                             // 0..63  (N tiles)
    const int lane = threadIdx.x;                          // 32 threads, EXEC all ones
    const int row = lane & 15;
    const int hi  = lane >> 4;

    const float* arow = rc + (bt * 16 + row) * L_ + 2 * hi;
    const float* brow = fc + (nt * 16 + row) * L_ + 2 * hi;

    v8f acc0 = {};
    v8f acc1 = {};
#pragma unroll 4
    for (int l0 = 0; l0 < L_; l0 += 8) {
        v2f a0 = *(const v2f*)(arow + l0);
        v2f b0 = *(const v2f*)(brow + l0);
        v2f a1 = *(const v2f*)(arow + l0 + 4);
        v2f b1 = *(const v2f*)(brow + l0 + 4);
        acc0 = __builtin_amdgcn_wmma_f32_16x16x4_f32(false, a0, false, b0,
                                                     (short)0, acc0, false, false);
        acc1 = __builtin_amdgcn_wmma_f32_16x16x4_f32(false, a1, false, b1,
                                                     (short)0, acc1, false, false);
    }
    v8f acc = acc0 + acc1;

    const int   ni = nt * 16 + row;
    const float nf = norm_fc[ni];
#pragma unroll
    for (int r = 0; r < 8; r++) {
        int bi = bt * 16 + r + 8 * hi;
        float den = nf * norm_rc[bi];
        float val = (den > 0.f) ? acc[r] / den : 0.f;
        corr[bi * NF_ + ni] = val;
    }
}

// ---------------------------------------------------------------------------
// Kernel 6: circular conv1d (k=3, pad=1 wrap) over channels {corr, dbg},
//           output transposed to [B, Nf, D]. Pure bandwidth: coalesced stores.
// ---------------------------------------------------------------------------
__global__ void k_conv(const float* __restrict__ corr, const float* __restrict__ dbg,
                       const float* __restrict__ w, float* __restrict__ out) {
    const int b  = blockIdx.y;                             // 0..63
    const int n0 = blockIdx.x * 16;                        // 0..1023 step 16
    const int tid = threadIdx.x;                           // 256 threads

    __shared__ float sc[18], sd[18];
    if (tid < 18) {
        int n = (n0 - 1 + tid) & (NF_ - 1);                // circular
        sc[tid] = corr[b * NF_ + n];
        sd[tid] = dbg[b * NF_ + n];
    }
    __syncthreads();

#pragma unroll
    for (int dslice = 0; dslice < 2; dslice++) {
        const int d = tid + dslice * 256;
        const float w0 = w[d * 6 + 0], w1 = w[d * 6 + 1], w2 = w[d * 6 + 2];
        const float w3 = w[d * 6 + 3], w4 = w[d * 6 + 4], w5 = w[d * 6 + 5];
#pragma unroll
        for (int i = 0; i < 16; i++) {
            float s = w0 * sc[i] + w1 * sc[i + 1] + w2 * sc[i + 2]
                    + w3 * sd[i] + w4 * sd[i + 1] + w5 * sd[i + 2];
            out[((size_t)b * NF_ + (size_t)(n0 + i)) * D_ + d] = s;
        }
    }
}

// ---------------------------------------------------------------------------
extern "C" void kernel_launch(void* const* d_in, const int* in_sizes, int n_in,
                              void* d_out, int out_size, void* d_ws, size_t ws_size,
                              hipStream_t stream) {
    (void)in_sizes; (void)n_in; (void)out_size; (void)ws_size;

    const float* x        = (const float*)d_in[0];   // [B, L, 1]
    const int*   features = (const int*)  d_in[1];   // [Nf, F]
    const float* conv_w   = (const float*)d_in[2];   // [D, 2, 3]
    float*       out      = (float*)d_out;           // [B, Nf, D]

    char* ws = (char*)d_ws;
    size_t off = 0;
    auto alloc = [&](size_t bytes) { char* p = ws + off; off = (off + bytes + 255) & ~(size_t)255; return p; };

    int*   tokens  = (int*)  alloc((size_t)B_ * L_ * 4);
    float* rc      = (float*)alloc((size_t)B_ * L_ * 4);
    float* fc      = (float*)alloc((size_t)NF_ * L_ * 4);
    float* norm_rc = (float*)alloc((size_t)B_ * 4);
    float* norm_fc = (float*)alloc((size_t)NF_ * 4);
    float* dist    = (float*)alloc((size_t)B_ * NF_ * 4);
    float* dbg     = (float*)alloc((size_t)B_ * NF_ * 4);
    float* corr    = (float*)alloc((size_t)B_ * NF_ * 4);

    k_tokens<<<dim3(B_), dim3(256), 0, stream>>>(x, tokens, rc, norm_rc);
    k_fc    <<<dim3(NF_), dim3(256), 0, stream>>>(features, fc, norm_fc);
    k_lev   <<<dim3(2, B_), dim3(256), 0, stream>>>(tokens, features, dist);
    k_dbg   <<<dim3(B_), dim3(256), 0, stream>>>(dist, dbg);
    k_corr  <<<dim3(B_ / 16, NF_ / 16), dim3(32), 0, stream>>>(rc, fc, norm_rc, norm_fc, corr);
    k_conv  <<<dim3(NF_ / 16, B_), dim3(256), 0, stream>>>(corr, dbg, conv_w, out);
}